// model_6597069767119
// MI455X (gfx1250) — compile-verified
//
#include <hip/hip_runtime.h>
#include <math.h>

// CDNA5 / gfx1250: wave32, WMMA 16x16x4 F32 (the only f32 WMMA shape, K=4 —
// exactly matches the homogeneous-coordinate affine transform here).
typedef __attribute__((ext_vector_type(2))) float v2f;
typedef __attribute__((ext_vector_type(8))) float v8f;

#define WAVE 32

// One wave per 16-point tile. For each 16-box tile we run 3 WMMAs
// (lx, ly, lz accumulators, each 16 points x 16 boxes), then do the
// containment test + running min-box-index in VALU.
__global__ __launch_bounds__(256) void pib_wmma_kernel(
    const float* __restrict__ points,   // [B, N, 3]
    const float* __restrict__ boxes,    // [B, M, 7] (cx,cy,cz,dx,dy,dz,rz)
    float* __restrict__ out,            // [B, N] first containing box idx or -1
    int B, int N, int M)
{
    const int lane          = threadIdx.x & (WAVE - 1);
    const int waveInBlock   = threadIdx.x >> 5;
    const int wavesPerBlock = blockDim.x >> 5;
    const int gwave         = blockIdx.x * wavesPerBlock + waveInBlock;
    const int tilesPerBatch = N >> 4;                 // 16 points per tile
    const int b             = gwave / tilesPerBatch;  // wave-uniform
    const int base          = (gwave % tilesPerBatch) << 4;
    if (b >= B) return;   // wave-uniform exit; EXEC stays all-1 for live waves

    const int n = lane & 15;   // matrix column (box) / point row index
    const int h = lane >> 4;   // lane half: selects K pair (K = v + 2h)

    // ---- A matrix (16x4 f32): row m = point n, K = (px, py, pz, 1) ----
    const float* p = points + ((size_t)b * N + base + n) * 3;
    const float px = p[0], py = p[1], pz = p[2];
    v2f a;
    a.x = h ? pz   : px;   // K = 2h
    a.y = h ? 1.0f : py;   // K = 2h + 1

    v8f best;
#pragma unroll
    for (int i = 0; i < 8; ++i) best[i] = 1e9f;

    const v8f zero = {};

    for (int t = 0; t < M; t += 16) {
        const int box = t + n;
        const float* bp = boxes + ((size_t)b * M + box) * 7;
        const float cx = bp[0], cy = bp[1], cz = bp[2];
        const float dx = bp[3], dy = bp[4], dz = bp[5], rz = bp[6];
        float s, c;
        __sincosf(rz, &s, &c);
        const float tx = -(cx * c + cy * s);   // lx constant term
        const float ty =  (cx * s - cy * c);   // ly constant term

        // ---- B matrices (4x16 f32): column n = box; VGPR v, half h -> K = v+2h
        // lx coeffs over K: ( c,  s, 0, tx )
        // ly coeffs over K: (-s,  c, 0, ty )
        // lz coeffs over K: ( 0,  0, 1, -cz)
        v2f blx, bly, blz;                 // ternaries -> v_cndmask (EXEC untouched)
        blx.x = h ? 0.0f : c;    blx.y = h ? tx  : s;
        bly.x = h ? 0.0f : -s;   bly.y = h ? ty  : c;
        blz.x = h ? 1.0f : 0.0f; blz.y = h ? -cz : 0.0f;

        // D = A x B : 16 points x 16 boxes, f32 accum (8 VGPRs)
        v8f lx = __builtin_amdgcn_wmma_f32_16x16x4_f32(
            false, a, false, blx, (short)0, zero, false, false);
        v8f ly = __builtin_amdgcn_wmma_f32_16x16x4_f32(
            false, a, false, bly, (short)0, zero, false, false);
        v8f lz = __builtin_amdgcn_wmma_f32_16x16x4_f32(
            false, a, false, blz, (short)0, zero, false, false);

        const float hx = dx * 0.5f, hy = dy * 0.5f, hz = dz * 0.5f;
        const float fbox = (float)box;
#pragma unroll
        for (int i = 0; i < 8; ++i) {
            // D element (VGPR i, this lane) = point (base + i + 8h), box (t + n)
            const bool in = (__builtin_fabsf(lx[i]) <= hx) &&
                            (__builtin_fabsf(ly[i]) <= hy) &&
                            (__builtin_fabsf(lz[i]) <= hz);
            const float cand = in ? fbox : 1e9f;
            best[i] = fminf(best[i], cand);
        }
    }

    // ---- min-index reduction across the 16 lanes of each half-wave ----
    // Row (point) for VGPR i lives in lanes [16h, 16h+16); xor masks < 16
    // never cross the half boundary.
#pragma unroll
    for (int i = 0; i < 8; ++i) {
        float v = best[i];
        v = fminf(v, __shfl_xor(v, 1));
        v = fminf(v, __shfl_xor(v, 2));
        v = fminf(v, __shfl_xor(v, 4));
        v = fminf(v, __shfl_xor(v, 8));
        best[i] = v;
    }

    // Lane with (lane%16)==i writes point (base + 8h + i): 16 stores per wave.
#pragma unroll
    for (int i = 0; i < 8; ++i) {
        if (n == i) {
            const float v = best[i];
            out[(size_t)b * N + base + (h << 3) + i] = (v >= 1e9f) ? -1.0f : v;
        }
    }
}

extern "C" void kernel_launch(void* const* d_in, const int* in_sizes, int n_in,
                              void* d_out, int out_size, void* d_ws, size_t ws_size,
                              hipStream_t stream) {
    const float* points = (const float*)d_in[0];   // [B, N, 3]
    const float* boxes  = (const float*)d_in[1];   // [B, M, 7]
    float* out = (float*)d_out;                    // [B, N] float32

    const int B = 4;
    const int N = in_sizes[0] / (3 * B);           // 65536
    const int M = in_sizes[1] / (7 * B);           // 128

    const int totalWaves    = B * (N / 16);        // one wave per 16-point tile
    const int block         = 256;                 // 8 waves / workgroup (wave32)
    const int wavesPerBlock = block / 32;
    const int grid          = (totalWaves + wavesPerBlock - 1) / wavesPerBlock;

    pib_wmma_kernel<<<grid, block, 0, stream>>>(points, boxes, out, B, N, M);
}